// E3SchNet_16561393893484
// MI455X (gfx1250) — compile-verified
//
#include <hip/hip_runtime.h>
#include <hip/hip_bf16.h>

#define N_NODES 10000
#define N_EDGES 160000
#define FDIM 128
#define NF 64
#define NRBF 32
#define NINT 2
#define EPB 32            // edges per block in k_edge

typedef __attribute__((ext_vector_type(2))) float v2f;
typedef __attribute__((ext_vector_type(8))) float v8f;
typedef __attribute__((ext_vector_type(4))) unsigned int u32x4;
typedef __attribute__((ext_vector_type(8))) int i32x8;
typedef __attribute__((ext_vector_type(4))) int i32x4;

__device__ __forceinline__ float ssp_act(float x) {
  // softplus(x) - log(2), overflow-safe
  float ax = fabsf(x);
  return fmaxf(x, 0.0f) + log1pf(__expf(-ax)) - 0.69314718056f;
}

// One 16x16 f32 tile: acc += A(16xK row-major, lda) * B(Kx16 row-major, ldb)
__device__ __forceinline__ v8f wmma_tile_f32(const float* __restrict__ A, int lda,
                                             const float* __restrict__ B, int ldb,
                                             int K, int lane, v8f acc) {
  const int l = lane & 15;
  const int hi2 = (lane >> 4) << 1;   // 0 or 2
  for (int k = 0; k < K; k += 4) {
    v2f av = { A[l * lda + k + hi2], A[l * lda + k + hi2 + 1] };
    v2f bv = { B[(k + hi2) * ldb + l], B[(k + hi2 + 1) * ldb + l] };
    acc = __builtin_amdgcn_wmma_f32_16x16x4_f32(false, av, false, bv, (short)0, acc,
                                                false, false);
  }
  return acc;
}

// TDM: DMA a 2D tile (rows x tile_cols f32, source row stride = tensor_cols) into LDS.
// D# built per CDNA5 ISA 8.3/8.4: group0 = count|lds_addr|global_addr|type=2,
// group1 = data_size=4B, dims/strides. Groups 2/3 zero (<=2D tensor). cpol=0.
__device__ __forceinline__ void tdm_load_2d(const float* gsrc, void* ldst,
                                            unsigned tile_cols, unsigned rows,
                                            unsigned tensor_cols) {
  unsigned long long ga = (unsigned long long)gsrc;
  unsigned lds = (unsigned)(unsigned long long)ldst;   // low 32 bits = LDS byte offset
  u32x4 g0;
  g0[0] = 1u;                                          // count=1 (valid), no gather
  g0[1] = lds;                                         // lds_addr
  g0[2] = (unsigned)(ga & 0xffffffffu);                // global_addr[31:0]
  g0[3] = (unsigned)((ga >> 32) & 0x01ffffffu) | (2u << 30);  // addr[56:32] | type=2
  i32x8 g1;
  g1[0] = (int)(2u << 16);            // workgroup_mask=0, data_size=2 (4 bytes)
  g1[1] = (int)(tensor_cols << 16);   // tensor_dim0[15:0]  @ bits[63:48]
  g1[2] = (int)(rows << 16);          // tensor_dim1[15:0]  @ bits[95:80]
  g1[3] = (int)(tile_cols << 16);     // tile_dim0          @ bits[127:112]
  g1[4] = (int)rows;                  // tile_dim1 @ [143:128], tile_dim2=0
  g1[5] = (int)tensor_cols;           // tensor_dim0_stride[31:0]
  g1[6] = 0;
  g1[7] = 0;
  i32x4 z4 = {0, 0, 0, 0};
  i32x8 z8 = {0, 0, 0, 0, 0, 0, 0, 0};
  __builtin_amdgcn_tensor_load_to_lds(g0, g1, z4, z4, z8, 0);
}

// xs = embed[species] @ W0   (gathered-A WMMA GEMM; 8 waves = 128 output cols)
__global__ __launch_bounds__(256) void k_embed(const float* __restrict__ emb,
                                               const int* __restrict__ species,
                                               const float* __restrict__ W0,
                                               float* __restrict__ xs) {
  const int lane = threadIdx.x & 31;
  const int wid  = threadIdx.x >> 5;
  const int row0 = blockIdx.x << 4;
  const int col0 = wid << 4;
  const int l = lane & 15, hi = lane >> 4, hi2 = hi << 1;
  const int sp = species[row0 + l];
  const float* Ar = emb + (size_t)sp * FDIM;
  v8f acc = {};
  for (int k = 0; k < FDIM; k += 4) {
    v2f av = { Ar[k + hi2], Ar[k + hi2 + 1] };
    v2f bv = { W0[(k + hi2) * FDIM + col0 + l], W0[(k + hi2 + 1) * FDIM + col0 + l] };
    acc = __builtin_amdgcn_wmma_f32_16x16x4_f32(false, av, false, bv, (short)0, acc,
                                                false, false);
  }
  #pragma unroll
  for (int v = 0; v < 8; ++v)
    xs[(size_t)(row0 + v + 8 * hi) * FDIM + col0 + l] = acc[v];
}

// C[MxN] = act(A[MxK] @ B[KxN] (+bias)) [optionally += into C]
__global__ __launch_bounds__(256) void k_gemm(const float* __restrict__ A,
                                              const float* __restrict__ B,
                                              const float* __restrict__ bias,
                                              float* __restrict__ C,
                                              int Kdim, int Ncols, int act, int accum) {
  const int lane = threadIdx.x & 31;
  const int wid  = threadIdx.x >> 5;
  const int row0 = blockIdx.x << 4;
  const int col0 = (blockIdx.y * 8 + wid) << 4;
  if (col0 >= Ncols) return;                   // wave-uniform
  v8f acc = {};
  acc = wmma_tile_f32(A + (size_t)row0 * Kdim, Kdim, B + col0, Ncols, Kdim, lane, acc);
  const int l = lane & 15, hi = lane >> 4;
  #pragma unroll
  for (int v = 0; v < 8; ++v) {
    float val = acc[v];
    if (bias) val += bias[col0 + l];
    if (act)  val = ssp_act(val);
    float* p = C + (size_t)(row0 + v + 8 * hi) * Ncols + col0 + l;
    *p = accum ? (*p + val) : val;
  }
}

// Fused edge pipeline, 32 edges/block, 512 threads (16 waves):
// TDM weight prefetch || geometry/RBF/gather  ->  WMMA MLP  ->  messages -> atomic segsum
__global__ __launch_bounds__(512) void k_edge(
    const float* __restrict__ pos, const int* __restrict__ send, const int* __restrict__ recv,
    const float* __restrict__ hs, const float* __restrict__ hv,      // hv: 3 planes of N*NF
    const float* __restrict__ W1, const float* __restrict__ b1,
    const float* __restrict__ W2, const float* __restrict__ b2,
    float* __restrict__ Ms, float* __restrict__ Mv)                  // Mv: 3 planes of N*FDIM
{
  __shared__ float sW1[NRBF][NF];        // 8 KB  (TDM-staged)
  __shared__ float sW2[NF][4 * NF];      // 64 KB (TDM-staged live block of W2)
  __shared__ float sF[EPB][NRBF];        // 4 KB
  __shared__ float sH[EPB][NF];          // 8 KB
  __shared__ float sWf[EPB][4 * NF];     // 32 KB
  __shared__ float sSj[EPB][NF];         // 8 KB
  __shared__ float sVj[3][EPB][NF];      // 24 KB
  __shared__ float sGeo[EPB][6];
  __shared__ int   sSend[EPB], sRecv[EPB];

  const int tid  = threadIdx.x;
  const int lane = tid & 31, wid = tid >> 5;
  const int e0   = blockIdx.x * EPB;

  // Wave 0: kick off async weight DMA (overlaps gather phase below).
  if (wid == 0) {
    tdm_load_2d(W1, &sW1[0][0], NF, NRBF, NF);          // 32x64, dense
    tdm_load_2d(W2, &sW2[0][0], 4 * NF, NF, 6 * NF);    // 64x256 tile of 64x384 tensor
  }

  if (tid < EPB) {
    int e = e0 + tid;
    int s = send[e], r = recv[e];
    sSend[tid] = s; sRecv[tid] = r;
    float rx = pos[r * 3 + 0] - pos[s * 3 + 0];
    float ry = pos[r * 3 + 1] - pos[s * 3 + 1];
    float rz = pos[r * 3 + 2] - pos[s * 3 + 2];
    float d  = sqrtf(rx * rx + ry * ry + rz * rz + 1e-12f);
    float inside = (d < 5.0f) ? 1.0f : 0.0f;
    float rcut = 0.5f * (__cosf(d * (3.14159265358979f / 5.0f)) + 1.0f) * inside;
    float sc = 1.73205080757f * inside / d;   // sqrt(3) * inside / d
    sGeo[tid][0] = rx * sc; sGeo[tid][1] = ry * sc; sGeo[tid][2] = rz * sc;
    sGeo[tid][3] = d;       sGeo[tid][4] = rcut;
  }
  __syncthreads();

  // RBF: centers = (k+1)*5/33, step = 5/33, *1.12
  for (int idx = tid; idx < EPB * NRBF; idx += 512) {
    int e = idx >> 5, k = idx & 31;
    float t = (sGeo[e][3] - (float)(k + 1) * (5.0f / 33.0f)) * (33.0f / 5.0f);
    sF[e][k] = __expf(-t * t) * 1.12f;
  }
  // Gather sender features (random access, L2-resident)
  for (int idx = tid; idx < EPB * NF; idx += 512) {
    int e = idx >> 6, f = idx & 63;
    int s = sSend[e];
    sSj[e][f]    = hs[(size_t)s * NF + f];
    sVj[0][e][f] = hv[(size_t)s * NF + f];
    sVj[1][e][f] = hv[(size_t)(N_NODES + s) * NF + f];
    sVj[2][e][f] = hv[(size_t)(2 * N_NODES + s) * NF + f];
  }
  if (wid == 0) __builtin_amdgcn_s_wait_tensorcnt(0);   // weights landed in LDS
  __syncthreads();

  // h = ssp(f_ij @ W1 + b1) : 32x64 = 8 WMMA tiles (waves 0..7), K=32, B from LDS
  if (wid < 8) {
    int rt = wid >> 2, col0 = (wid & 3) << 4;
    v8f acc = {};
    acc = wmma_tile_f32(&sF[rt * 16][0], NRBF, &sW1[0][col0], NF, NRBF, lane, acc);
    int l = lane & 15, hi = lane >> 4;
    #pragma unroll
    for (int v = 0; v < 8; ++v)
      sH[rt * 16 + v + 8 * hi][col0 + l] = ssp_act(acc[v] + b1[col0 + l]);
  }
  __syncthreads();

  // Wf = (h @ W2[:, :256] + b2) * rcut : 32x256 = 32 tiles over 16 waves, B from LDS
  for (int t = wid; t < 32; t += 16) {
    int rt = t >> 4, col0 = (t & 15) << 4;
    v8f acc = {};
    acc = wmma_tile_f32(&sH[rt * 16][0], NF, &sW2[0][col0], 4 * NF, NF, lane, acc);
    int l = lane & 15, hi = lane >> 4;
    #pragma unroll
    for (int v = 0; v < 8; ++v) {
      int row = rt * 16 + v + 8 * hi;
      sWf[row][col0 + l] = (acc[v] + b2[col0 + l]) * sGeo[row][4];
    }
  }
  __syncthreads();

  // Message formation + fp32 atomic segment-sum into Ms (N x 128) / Mv (3 planes N x 128)
  for (int idx = tid; idx < EPB * NF; idx += 512) {
    int e = idx >> 6, f = idx & 63;
    int r = sRecv[e];
    float b0 = sGeo[e][0], b1v = sGeo[e][1], b2v = sGeo[e][2];
    float sj = sSj[e][f];
    float v0 = sVj[0][e][f], v1 = sVj[1][e][f], v2 = sVj[2][e][f];
    float ws1 = sWf[e][f],          ws2 = sWf[e][NF + f];
    float wv1 = sWf[e][2 * NF + f], wv2 = sWf[e][3 * NF + f];
    float os2 = (v0 * b0 + v1 * b1v + v2 * b2v) * 0.57735026919f;  // 1/sqrt(3)
    float* MsR = Ms + (size_t)r * FDIM;
    atomicAdd(MsR + f,       sj * ws1);
    atomicAdd(MsR + NF + f,  os2 * ws2);
    float sw = sj * wv1;
    float* Mv0 = Mv + (size_t)r * FDIM;
    float* Mv1 = Mv + (size_t)(N_NODES + r) * FDIM;
    float* Mv2 = Mv + (size_t)(2 * N_NODES + r) * FDIM;
    atomicAdd(Mv0 + f,      sw * b0);
    atomicAdd(Mv0 + NF + f, v0 * wv2);
    atomicAdd(Mv1 + f,      sw * b1v);
    atomicAdd(Mv1 + NF + f, v1 * wv2);
    atomicAdd(Mv2 + f,      sw * b2v);
    atomicAdd(Mv2 + NF + f, v2 * wv2);
  }
}

// out[n] = concat(xs[n, 0:128], xv[n, f, c] interleaved f-major)
__global__ __launch_bounds__(256) void k_output(const float* __restrict__ xs,
                                                const float* __restrict__ xv,
                                                float* __restrict__ out) {
  int idx = blockIdx.x * blockDim.x + threadIdx.x;
  const int total = N_NODES * 512;
  if (idx >= total) return;
  int n = idx >> 9, j = idx & 511;
  float v;
  if (j < FDIM) {
    v = xs[(size_t)n * FDIM + j];
  } else {
    int jj = j - FDIM;
    int f = jj / 3, c = jj - 3 * f;
    v = xv[(size_t)(c * N_NODES + n) * FDIM + f];
  }
  out[idx] = v;
}

extern "C" void kernel_launch(void* const* d_in, const int* in_sizes, int n_in,
                              void* d_out, int out_size, void* d_ws, size_t ws_size,
                              hipStream_t stream) {
  (void)in_sizes; (void)n_in; (void)out_size; (void)ws_size;
  const float* positions = (const float*)d_in[0];
  const int*   species   = (const int*)d_in[1];
  const int*   senders   = (const int*)d_in[2];
  const int*   receivers = (const int*)d_in[3];
  const float* embed     = (const float*)d_in[4];
  const float* W0        = (const float*)d_in[5];
  const float* Win_s     = (const float*)d_in[6];
  const float* Win_v     = (const float*)d_in[7];
  const float* W1        = (const float*)d_in[8];
  const float* b1        = (const float*)d_in[9];
  const float* W2        = (const float*)d_in[10];
  const float* b2        = (const float*)d_in[11];
  const float* Wo1_s     = (const float*)d_in[12];
  const float* Wo1_v     = (const float*)d_in[13];
  const float* Wo2_s     = (const float*)d_in[14];
  const float* Wo2_v     = (const float*)d_in[15];
  float* out = (float*)d_out;

  // Workspace carve-out (~72 MB, L2-resident on MI455X's 192 MB L2)
  float* xs   = (float*)d_ws;                         // N*128
  float* xv   = xs   + (size_t)N_NODES * FDIM;        // 3 planes N*128
  float* hs   = xv   + (size_t)3 * N_NODES * FDIM;    // N*64
  float* hv   = hs   + (size_t)N_NODES * NF;          // 3 planes N*64
  float* Ms   = hv   + (size_t)3 * N_NODES * NF;      // N*128
  float* Mv   = Ms   + (size_t)N_NODES * FDIM;        // 3 planes N*128 (contiguous after Ms)
  float* tmp  = Mv   + (size_t)3 * N_NODES * FDIM;    // N*128
  float* tmpv = tmp  + (size_t)N_NODES * FDIM;        // 3 planes N*128

  (void)hipMemsetAsync(xv, 0, (size_t)3 * N_NODES * FDIM * sizeof(float), stream);
  k_embed<<<N_NODES / 16, 256, 0, stream>>>(embed, species, W0, xs);

  for (int i = 0; i < NINT; ++i) {
    // Input projections: hs = xs@Win_s[i], hv_c = xv_c@Win_v[i]
    k_gemm<<<dim3(N_NODES / 16, 1), 256, 0, stream>>>(
        xs, Win_s + (size_t)i * FDIM * NF, nullptr, hs, FDIM, NF, 0, 0);
    for (int c = 0; c < 3; ++c)
      k_gemm<<<dim3(N_NODES / 16, 1), 256, 0, stream>>>(
          xv + (size_t)c * N_NODES * FDIM, Win_v + (size_t)i * FDIM * NF, nullptr,
          hv + (size_t)c * N_NODES * NF, FDIM, NF, 0, 0);

    // Zero Ms|Mv (contiguous 4 planes of N*128), then fused edge pass
    (void)hipMemsetAsync(Ms, 0, (size_t)4 * N_NODES * FDIM * sizeof(float), stream);
    k_edge<<<N_EDGES / EPB, 512, 0, stream>>>(
        positions, senders, receivers, hs, hv,
        W1 + (size_t)i * NRBF * NF, b1 + (size_t)i * NF,
        W2 + (size_t)i * NF * 6 * NF, b2 + (size_t)i * 6 * NF, Ms, Mv);

    // Node updates: xs += ssp(Ms@Wo1_s)@Wo2_s ; xv_c += (Mv_c@Wo1_v)@Wo2_v
    k_gemm<<<dim3(N_NODES / 16, 1), 256, 0, stream>>>(
        Ms, Wo1_s + (size_t)i * FDIM * FDIM, nullptr, tmp, FDIM, FDIM, 1, 0);
    k_gemm<<<dim3(N_NODES / 16, 1), 256, 0, stream>>>(
        tmp, Wo2_s + (size_t)i * FDIM * FDIM, nullptr, xs, FDIM, FDIM, 0, 1);
    for (int c = 0; c < 3; ++c) {
      k_gemm<<<dim3(N_NODES / 16, 1), 256, 0, stream>>>(
          Mv + (size_t)c * N_NODES * FDIM, Wo1_v + (size_t)i * FDIM * FDIM, nullptr,
          tmpv + (size_t)c * N_NODES * FDIM, FDIM, FDIM, 0, 0);
      k_gemm<<<dim3(N_NODES / 16, 1), 256, 0, stream>>>(
          tmpv + (size_t)c * N_NODES * FDIM, Wo2_v + (size_t)i * FDIM * FDIM, nullptr,
          xv + (size_t)c * N_NODES * FDIM, FDIM, FDIM, 0, 1);
    }
  }

  const int total = N_NODES * 512;
  k_output<<<(total + 255) / 256, 256, 0, stream>>>(xs, xv, out);
}